// Model_68874095559233
// MI455X (gfx1250) — compile-verified
//
#include <hip/hip_runtime.h>
#include <math.h>

// psi[k] = prod_{j=0..24} ( bit_j(k) ? cos(theta[24-j]) : sin(theta[24-j]) )
// out    = -(psi . omega),  |omega| = 2^25
//
// k = b*2^15 + it*2^10 + t*2^2 + v   (separable product across bit groups)
//   b  = blockIdx.x  (10 bits, positions 15..24)
//   it = iteration   ( 5 bits, positions 10..14)
//   t  = threadIdx.x ( 8 bits, positions  2..9 )
//   v  = float4 lane ( 2 bits, positions  0..1 )

typedef float v2f __attribute__((ext_vector_type(2)));
typedef float v8f __attribute__((ext_vector_type(8)));

#define NBLK 1024
#define NTHR 256
#define NIT  32

// d_ws float layout
#define TAB_S    0     // sin(theta[0..24])
#define TAB_C    32    // cos(theta[0..24])
#define PART_OFF 64    // 1024 block partials

// ---- setup: 25 accurate sincos evaluations, done once by one wave ----
__global__ void setup_tables(const float* __restrict__ theta,
                             float* __restrict__ ws) {
    const int i = threadIdx.x;
    if (i < 25) {
        float s, c;
        sincosf(theta[i], &s, &c);
        ws[TAB_S + i] = s;
        ws[TAB_C + i] = c;
    }
}

// Sum 32 lane partials of a wave32 with V_WMMA_F32_16X16X4_F32.
// A (16x4): lanes 0-15 feed column K=0, lanes 16-31 feed column K=2; B = ones.
// D[m][n] = p[m] + p[m+16]; summing the 8 D VGPRs gives one half per lane
// group, shfl_xor(16) combines halves -> every lane holds the wave total.
__device__ __forceinline__ float wave_reduce_wmma(float v) {
    v2f a; a.x = v;    a.y = 0.0f;
    v2f b; b.x = 1.0f; b.y = 1.0f;
    v8f c = {};
    v8f d = __builtin_amdgcn_wmma_f32_16x16x4_f32(
        false, a, false, b, (short)0, c, false, false);
    float s = ((d[0] + d[1]) + (d[2] + d[3])) + ((d[4] + d[5]) + (d[6] + d[7]));
    s += __shfl_xor(s, 16, 32);
    return s;
}

__global__ __launch_bounds__(NTHR)
void kron_dot_kernel(const float* __restrict__ omega,
                     const float* __restrict__ tab,     // d_ws: sin/cos tables
                     float* __restrict__ partials) {    // d_ws + PART_OFF
    const int t = threadIdx.x;
    const int b = blockIdx.x;

    // P_block: bits 15..24 from b (bit q of b -> theta[9-q]); wave-uniform.
    float pb = 1.0f;
#pragma unroll
    for (int q = 0; q < 10; ++q)
        pb *= ((b >> q) & 1) ? tab[TAB_C + 9 - q] : tab[TAB_S + 9 - q];

    // P_thread: bits 2..9 from t (bit q of t -> theta[22-q]); per-lane selects.
    float pt = 1.0f;
#pragma unroll
    for (int q = 0; q < 8; ++q)
        pt *= ((t >> q) & 1) ? tab[TAB_C + 22 - q] : tab[TAB_S + 22 - q];

    const float scale = pb * pt;

    // Iteration-bit factors: bits 10..14 (bit q of it -> theta[14-q]).
    float sit[5], cit[5];
#pragma unroll
    for (int q = 0; q < 5; ++q) {
        sit[q] = tab[TAB_S + 14 - q];
        cit[q] = tab[TAB_C + 14 - q];
    }

    // Vector-lane factors: bit0 -> theta[24], bit1 -> theta[23].
    const float s24 = tab[TAB_S + 24], c24 = tab[TAB_C + 24];
    const float s23 = tab[TAB_S + 23], c23 = tab[TAB_C + 23];
    const float pv0 = s24 * s23, pv1 = c24 * s23, pv2 = s24 * c23, pv3 = c24 * c23;

    // Streaming loop: 32 x global_load_b128 per thread off one base register,
    // constant instruction offsets (it*4096 B fits signed 24-bit IOFFSET).
    const float4* __restrict__ om4 = (const float4*)omega;
    const size_t base = (size_t)b * 8192 + (size_t)t;   // float4 units

    float a0 = 0.f, a1 = 0.f, a2 = 0.f, a3 = 0.f;
#pragma unroll
    for (int it = 0; it < NIT; ++it) {
        float p = scale;
        p *= (it & 1)  ? cit[0] : sit[0];
        p *= (it & 2)  ? cit[1] : sit[1];
        p *= (it & 4)  ? cit[2] : sit[2];
        p *= (it & 8)  ? cit[3] : sit[3];
        p *= (it & 16) ? cit[4] : sit[4];
        float4 w = om4[base + (size_t)it * NTHR];
        a0 = fmaf(w.x, p, a0);
        a1 = fmaf(w.y, p, a1);
        a2 = fmaf(w.z, p, a2);
        a3 = fmaf(w.w, p, a3);
    }
    float part = (a0 * pv0 + a1 * pv1) + (a2 * pv2 + a3 * pv3);

    // Block reduction: WMMA wave reduce (EXEC all ones here), LDS across 8 waves.
    float ws = wave_reduce_wmma(part);

    __shared__ float smem[NTHR / 32];
    const int wave = t >> 5, lane = t & 31;
    if (lane == 0) smem[wave] = ws;
    __syncthreads();
    if (t == 0) {
        float s = 0.f;
#pragma unroll
        for (int w = 0; w < NTHR / 32; ++w) s += smem[w];
        partials[b] = s;
    }
}

__global__ __launch_bounds__(NTHR)
void final_reduce_kernel(const float* __restrict__ partials,
                         float* __restrict__ out) {
    const int t = threadIdx.x;
    const float4* p4 = (const float4*)partials;   // 1024 = 256 x float4
    float4 v = p4[t];
    float s = (v.x + v.y) + (v.z + v.w);

    float ws = wave_reduce_wmma(s);               // all threads active

    __shared__ float smem[NTHR / 32];
    if ((t & 31) == 0) smem[t >> 5] = ws;
    __syncthreads();
    if (t == 0) {
        float tot = 0.f;
#pragma unroll
        for (int w = 0; w < NTHR / 32; ++w) tot += smem[w];
        out[0] = -tot;                            // reference returns -(psi @ omega)
    }
}

extern "C" void kernel_launch(void* const* d_in, const int* in_sizes, int n_in,
                              void* d_out, int out_size, void* d_ws, size_t ws_size,
                              hipStream_t stream) {
    const float* omega = (const float*)d_in[0];   // 2^25 f32
    const float* theta = (const float*)d_in[1];   // 25 f32
    float* ws  = (float*)d_ws;
    float* out = (float*)d_out;

    setup_tables<<<1, 32, 0, stream>>>(theta, ws);
    kron_dot_kernel<<<NBLK, NTHR, 0, stream>>>(omega, ws, ws + PART_OFF);
    final_reduce_kernel<<<1, NTHR, 0, stream>>>(ws + PART_OFF, out);
}